// SDDH_27556510171208
// MI455X (gfx1250) — compile-verified
//
#include <hip/hip_runtime.h>
#include <math.h>

#define DIMS   128
#define KS     3
#define NPOS   8
#define BATCH  4
#define H_     240
#define W_     320
#define NK     4000
#define HW     (H_*W_)
#define TOTAL_KP (BATCH*NK)       /* 16000 */
#define MCOLS    (TOTAL_KP*NPOS)  /* 128000 columns of feats */

#define F_BYTES   ((size_t)DIMS * MCOLS * 4)          /* 64 MiB  */
#define XT_BYTES  ((size_t)BATCH * HW * DIMS * 4)     /* 150 MiB */

typedef float v2f __attribute__((ext_vector_type(2)));
typedef float v4f __attribute__((ext_vector_type(4)));
typedef float v8f __attribute__((ext_vector_type(8)));

// ---------------------------------------------------------------------------
// Kernel 0: transpose x [B,C,H,W] -> xT [B,H*W,C] via 32x33 LDS tile.
// Both global read (along hw) and write (along c) are fully coalesced.
// ---------------------------------------------------------------------------
__global__ __launch_bounds__(256) void sddh_transpose(
    const float* __restrict__ x, float* __restrict__ xT)
{
  __shared__ float s[32][33];
  const int hw0 = blockIdx.x * 32;
  const int c0  = blockIdx.y * 32;
  const int b   = blockIdx.z;
  const int col = threadIdx.x & 31;
  const int rb  = threadIdx.x >> 5;          // 0..7

  const float* src = x + (size_t)b * DIMS * HW;
  float*       dst = xT + (size_t)b * HW * DIMS;

#pragma unroll
  for (int r = 0; r < 4; ++r) {              // read: rows = channels
    const int row = rb + r * 8;
    s[row][col] = src[(size_t)(c0 + row) * HW + hw0 + col];
  }
  __syncthreads();
#pragma unroll
  for (int r = 0; r < 4; ++r) {              // write: rows = pixels
    const int row = rb + r * 8;
    dst[(size_t)(hw0 + row) * DIMS + c0 + col] = s[col][row];
  }
}

// ---------------------------------------------------------------------------
// Kernel 1: per-keypoint offset MLP + bilinear gather -> F[c][m], m=kp*8+p
// One wave32 per keypoint; each lane owns 4 channels.
// CHW=true : read x  in [C,H,W]  (scattered per-channel gathers, fallback)
// CHW=false: read xT in [H,W,C]  (one coalesced 512B wave-load per tap)
// ---------------------------------------------------------------------------
template <bool CHW>
__global__ __launch_bounds__(256) void sddh_gather(
    const float* __restrict__ xsrc, const float* __restrict__ kpts,
    const float* __restrict__ w1, const float* __restrict__ b1,
    const float* __restrict__ w2, const float* __restrict__ b2,
    float* __restrict__ F)
{
  const int wave = threadIdx.x >> 5;
  const int lane = threadIdx.x & 31;
  const int kp   = blockIdx.x * 8 + wave;           // 0..15999
  const int b    = kp / NK;
  const float* xi = xsrc + (size_t)b * (DIMS * HW);

  const float kx = kpts[kp * 2 + 0], ky = kpts[kp * 2 + 1];
  const float kwx = (kx * 0.5f + 0.5f) * (float)(W_ - 1);
  const float kwy = (ky * 0.5f + 0.5f) * (float)(H_ - 1);
  const int kwlx = (int)kwx;                        // trunc == floor (kw >= 0)
  const int kwly = (int)kwy;
  int cx = (int)((float)kwlx - 0.5f);               // (kwl - KS/2 + 1) trunc
  int cy = (int)((float)kwly - 0.5f);
  cx = min(max(cx, 0), W_ - 1 - KS);
  cy = min(max(cy, 0), H_ - 1 - KS);

  // ---- 3x3 patch for my 4 channels ----
  const int c0 = lane * 4;
  float patch[4][9];
  if (CHW) {
#pragma unroll
    for (int cc = 0; cc < 4; ++cc) {
      const float* base = xi + (size_t)(c0 + cc) * HW + cy * W_ + cx;
#pragma unroll
      for (int i = 0; i < 3; ++i)
#pragma unroll
        for (int j = 0; j < 3; ++j)
          patch[cc][i * 3 + j] = base[i * W_ + j];
    }
  } else {
#pragma unroll
    for (int i = 0; i < 3; ++i)
#pragma unroll
      for (int j = 0; j < 3; ++j) {
        const v4f pv = *(const v4f*)(xi + (size_t)((cy + i) * W_ + cx + j) * DIMS + c0);
        const int t = i * 3 + j;
        patch[0][t] = pv.x; patch[1][t] = pv.y;
        patch[2][t] = pv.z; patch[3][t] = pv.w;
      }
  }

  // ---- offset conv (full 3x3 contraction), partial over my channels ----
  float part[16];
#pragma unroll
  for (int o = 0; o < 16; ++o) {
    float acc = 0.f;
#pragma unroll
    for (int cc = 0; cc < 4; ++cc) {
      const float* wr = w1 + ((size_t)o * DIMS + (c0 + cc)) * 9;
#pragma unroll
      for (int t = 0; t < 9; ++t) acc += patch[cc][t] * wr[t];
    }
    part[o] = acc;
  }
  // butterfly reduce across the wave32 -> all lanes hold full sums
#pragma unroll
  for (int off = 16; off >= 1; off >>= 1)
#pragma unroll
    for (int o = 0; o < 16; ++o)
      part[o] += __shfl_xor(part[o], off, 32);

  float o1[16];
#pragma unroll
  for (int o = 0; o < 16; ++o) o1[o] = fmaxf(part[o] + b1[o], 0.f);

  const float maxoff = 80.f;                        // max(H,W)/4
  float o2[16];
#pragma unroll
  for (int j = 0; j < 16; ++j) {
    float a = b2[j];
#pragma unroll
    for (int k = 0; k < 16; ++k) a += w2[j * 16 + k] * o1[k];
    o2[j] = fminf(fmaxf(a, -maxoff), maxoff);
  }

  // ---- bilinear sample 8 positions x my 4 channels (zeros padding) ----
  float feat[4][NPOS];
#pragma unroll
  for (int p = 0; p < NPOS; ++p) {
    const float px = kwx + o2[p];
    const float py = kwy + o2[NPOS + p];
    const float x0f = floorf(px), y0f = floorf(py);
    const float wx = px - x0f, wy = py - y0f;
    int   idx[4];
    float wgt[4];
#pragma unroll
    for (int t = 0; t < 4; ++t) {
      const float xc = x0f + (float)(t & 1);
      const float yc = y0f + (float)(t >> 1);
      const bool valid = (xc >= 0.f) && (xc <= (float)(W_ - 1)) &&
                         (yc >= 0.f) && (yc <= (float)(H_ - 1));
      const int xic = (int)fminf(fmaxf(xc, 0.f), (float)(W_ - 1));
      const int yic = (int)fminf(fmaxf(yc, 0.f), (float)(H_ - 1));
      idx[t] = valid ? (yic * W_ + xic) : -1;
      const float fx = (t & 1)  ? wx : (1.f - wx);
      const float fy = (t >> 1) ? wy : (1.f - wy);
      wgt[t] = fx * fy;
    }
    if (CHW) {
#pragma unroll
      for (int cc = 0; cc < 4; ++cc) {
        const float* ch = xi + (size_t)(c0 + cc) * HW;
        float acc = 0.f;
#pragma unroll
        for (int t = 0; t < 4; ++t)
          acc += (idx[t] >= 0 ? ch[idx[t]] : 0.f) * wgt[t];
        feat[cc][p] = acc;
      }
    } else {
      v4f acc = {};
#pragma unroll
      for (int t = 0; t < 4; ++t) {
        const v4f zero = {};
        const v4f v = (idx[t] >= 0)
            ? *(const v4f*)(xi + (size_t)idx[t] * DIMS + c0) : zero;
        const v4f wv = {wgt[t], wgt[t], wgt[t], wgt[t]};
        acc += v * wv;
      }
      feat[0][p] = acc.x; feat[1][p] = acc.y;
      feat[2][p] = acc.z; feat[3][p] = acc.w;
    }
  }
  // coalesced-ish stores: two float4 per channel (kp*8 floats is 32B aligned)
#pragma unroll
  for (int cc = 0; cc < 4; ++cc) {
    float* dst = F + (size_t)(c0 + cc) * MCOLS + (size_t)kp * NPOS;
    v4f lo = {feat[cc][0], feat[cc][1], feat[cc][2], feat[cc][3]};
    v4f hi = {feat[cc][4], feat[cc][5], feat[cc][6], feat[cc][7]};
    *(v4f*)(dst)     = lo;
    *(v4f*)(dst + 4) = hi;
  }
}

// ---------------------------------------------------------------------------
// Kernel 2: sf_conv GEMM (128x128 @ 128xM) + SELU, IN PLACE on F.
// One wave32 per 16-column tile; 8 output d-tiles; WMMA f32 16x16x4.
// All loads of the tile precede all stores, columns disjoint across waves.
// ---------------------------------------------------------------------------
__global__ __launch_bounds__(256) void sddh_sfconv(
    const float* __restrict__ sf_w, float* __restrict__ F)
{
  const int wave = threadIdx.x >> 5;
  const int lane = threadIdx.x & 31;
  const int tile = blockIdx.x * 8 + wave;           // 0..7999
  const int m0   = tile * 16;
  const int half = lane >> 4;                       // 0 | 1
  const int l15  = lane & 15;
  const int koff = half * 2;                        // K split 0,1 / 2,3

  v8f acc[8] = {};

  for (int k0 = 0; k0 < DIMS; k0 += 4) {
    v2f bf;   // B[k, n] = F[c=k][m0+n], rows contiguous in m -> coalesced
    bf.x = F[(size_t)(k0 + koff)     * MCOLS + m0 + l15];
    bf.y = F[(size_t)(k0 + koff + 1) * MCOLS + m0 + l15];
#pragma unroll
    for (int dt = 0; dt < 8; ++dt) {
      v2f af;  // A[m=d_local, k=c] = sf_w[d][c]
      af.x = sf_w[(dt * 16 + l15) * DIMS + k0 + koff];
      af.y = sf_w[(dt * 16 + l15) * DIMS + k0 + koff + 1];
      acc[dt] = __builtin_amdgcn_wmma_f32_16x16x4_f32(
          false, af, false, bf, (short)0, acc[dt], false, false);
    }
  }

  const float SC = 1.0507009873554805f, AL = 1.6732632423543772f;
#pragma unroll
  for (int dt = 0; dt < 8; ++dt) {
#pragma unroll
    for (int r = 0; r < 8; ++r) {
      float v = acc[dt][r];
      v = (v > 0.f) ? (SC * v) : (SC * AL * expm1f(v));  // SELU
      const int d = dt * 16 + r + half * 8;              // C/D layout
      F[(size_t)d * MCOLS + m0 + l15] = v;
    }
  }
}

// ---------------------------------------------------------------------------
// Kernel 3: agg contraction as 16x128x1024 GEMM per 16-keypoint tile
// (K = p*128+c folded; agg_w[p][c][d] is exactly B[k][d] row-major),
// then L2-normalize each 128-d descriptor.
// ---------------------------------------------------------------------------
__global__ __launch_bounds__(256) void sddh_agg(
    const float* __restrict__ F, const float* __restrict__ aggw,
    float* __restrict__ out)
{
  const int wave = threadIdx.x >> 5;
  const int lane = threadIdx.x & 31;
  const int tile = blockIdx.x * 8 + wave;           // 0..999
  const int kp0  = tile * 16;
  const int half = lane >> 4;
  const int l15  = lane & 15;
  const int koff = half * 2;

  v8f acc[8] = {};

  for (int k0 = 0; k0 < NPOS * DIMS; k0 += 4) {     // K = 1024
    const int ka = k0 + koff;                       // 4-group never crosses c=128
    v2f af;                                         // A[m=kp_local, k] = G[c][(kp0+m)*8+p]
    {
      const int p = ka >> 7, c = ka & 127;
      af.x = F[(size_t)c * MCOLS + (kp0 + l15) * NPOS + p];
    }
    {
      const int kb = ka + 1;
      const int p = kb >> 7, c = kb & 127;
      af.y = F[(size_t)c * MCOLS + (kp0 + l15) * NPOS + p];
    }
#pragma unroll
    for (int dt = 0; dt < 8; ++dt) {
      v2f bf;                                       // B[k][d] = aggw[k*128 + d]
      bf.x = aggw[(size_t)ka       * DIMS + dt * 16 + l15];
      bf.y = aggw[(size_t)(ka + 1) * DIMS + dt * 16 + l15];
      acc[dt] = __builtin_amdgcn_wmma_f32_16x16x4_f32(
          false, af, false, bf, (short)0, acc[dt], false, false);
    }
  }

  // L2 norm over d: each lane holds 64 of the 128 d-values for column l15;
  // its partner lane (lane^16) holds the other 64.
  float ssq = 0.f;
#pragma unroll
  for (int dt = 0; dt < 8; ++dt)
#pragma unroll
    for (int r = 0; r < 8; ++r) ssq += acc[dt][r] * acc[dt][r];
  ssq += __shfl_xor(ssq, 16, 32);
  const float inv = 1.f / fmaxf(sqrtf(ssq), 1e-12f);

#pragma unroll
  for (int dt = 0; dt < 8; ++dt)
#pragma unroll
    for (int r = 0; r < 8; ++r) {
      const int d = dt * 16 + r + half * 8;
      out[(size_t)(kp0 + l15) * DIMS + d] = acc[dt][r] * inv;
    }
}

// ---------------------------------------------------------------------------
extern "C" void kernel_launch(void* const* d_in, const int* in_sizes, int n_in,
                              void* d_out, int out_size, void* d_ws, size_t ws_size,
                              hipStream_t stream) {
  const float* x    = (const float*)d_in[0];
  const float* kpts = (const float*)d_in[1];
  const float* w1   = (const float*)d_in[2];
  const float* b1   = (const float*)d_in[3];
  const float* w2   = (const float*)d_in[4];
  const float* b2   = (const float*)d_in[5];
  const float* sf_w = (const float*)d_in[6];
  const float* aggw = (const float*)d_in[7];
  float* out = (float*)d_out;
  float* F   = (float*)d_ws;                 // 64 MiB feats buffer at ws+0

  (void)in_sizes; (void)n_in; (void)out_size;

  if (ws_size >= F_BYTES + XT_BYTES) {
    // Fast path: transpose x -> [B,H,W,C] so every keypoint tap is one
    // coalesced 512B wave-load, then gather from the transposed map.
    float* xT = (float*)((char*)d_ws + F_BYTES);
    sddh_transpose<<<dim3(HW / 32, DIMS / 32, BATCH), 256, 0, stream>>>(x, xT);
    sddh_gather<false><<<TOTAL_KP / 8, 256, 0, stream>>>(xT, kpts, w1, b1, w2, b2, F);
  } else {
    // Fallback: gather straight from [B,C,H,W] (scattered).
    sddh_gather<true><<<TOTAL_KP / 8, 256, 0, stream>>>(x, kpts, w1, b1, w2, b2, F);
  }
  sddh_sfconv<<<(MCOLS / 16) / 8, 256, 0, stream>>>(sf_w, F);
  sddh_agg<<<(TOTAL_KP / 16) / 8, 256, 0, stream>>>(F, aggw, out);
}